// GTLayer_83133386981889
// MI455X (gfx1250) — compile-verified
//
#include <hip/hip_runtime.h>
#include <hip/hip_bf16.h>

#define LATDIM 128
#define HEADS  8

typedef __attribute__((ext_vector_type(2))) float v2f;
typedef __attribute__((ext_vector_type(8))) float v8f;

// ---------------------------------------------------------------------------
// Kernel 0: zero the scatter-add accumulators (d_out resEmbeds + norm ws).
// ---------------------------------------------------------------------------
__global__ void zero_accum_kernel(float* __restrict__ res, float* __restrict__ norm,
                                  int nRes, int nNorm) {
    int i = blockIdx.x * blockDim.x + threadIdx.x;
    int stride = gridDim.x * blockDim.x;
    for (int j = i; j < nRes; j += stride)  res[j]  = 0.0f;
    for (int j = i; j < nNorm; j += stride) norm[j] = 0.0f;
}

// ---------------------------------------------------------------------------
// Kernel 1: Q/K/V = embeds @ W{q,k,v}  (per-NODE projection; gather commutes
// with the linear map, cutting GEMM FLOPs 16x vs the per-edge reference).
// One wave per 16x16 output tile, K-loop of 32 x V_WMMA_F32_16X16X4_F32.
// blockDim = 256 (8 waves = 8 N-tiles covering all 128 output columns).
// grid = (N/16, 3): blockIdx.y selects Wq/Wk/Wv. N must be a multiple of 16
// (50000 = 3125*16) so EXEC is all-ones at every WMMA.
// ---------------------------------------------------------------------------
__global__ __launch_bounds__(256) void qkv_gemm_kernel(
    const float* __restrict__ emb,
    const float* __restrict__ Wq, const float* __restrict__ Wk,
    const float* __restrict__ Wv,
    float* __restrict__ out, int N) {

    __shared__ float ldsA[16 * 132];   // 16x128 A panel, padded stride 132

    const int mtile = blockIdx.x;
    const int mat   = blockIdx.y;
    const float* __restrict__ W = (mat == 0) ? Wq : (mat == 1 ? Wk : Wv);
    float* __restrict__ O = out + (size_t)mat * N * LATDIM;

    // Cooperative stage of the 16x128 A panel: 256 threads x 8 floats.
    {
        const int t   = threadIdx.x;
        const int elt = t * 8;
        const int row = elt >> 7;      // /128
        const int col = elt & 127;
        const float4* src = (const float4*)(emb + (size_t)mtile * 16 * LATDIM);
        float4 x0 = src[elt / 4];
        float4 x1 = src[elt / 4 + 1];
        *(float4*)&ldsA[row * 132 + col]     = x0;
        *(float4*)&ldsA[row * 132 + col + 4] = x1;
    }
    __syncthreads();

    const int lane = threadIdx.x & 31;
    const int wv   = threadIdx.x >> 5;      // N-tile 0..7
    const int l15  = lane & 15;             // A: M index / B: N index
    const int koff = (lane >> 4) << 1;      // 0 for lanes 0-15, 2 for 16-31
    const int ncol = wv * 16 + l15;

    v8f acc = {0.f, 0.f, 0.f, 0.f, 0.f, 0.f, 0.f, 0.f};

#pragma unroll 4
    for (int k0 = 0; k0 < LATDIM; k0 += 4) {
        // A 16x4 tile: lane<16 -> (M=l15, K=k0,k0+1); lane>=16 -> K=k0+2,k0+3
        v2f a = *(const v2f*)&ldsA[l15 * 132 + k0 + koff];
        // B 4x16 tile (row-major W[k][n]): same half-wave K split, N=l15
        v2f b;
        b.x = W[(k0 + koff)     * LATDIM + ncol];
        b.y = W[(k0 + koff + 1) * LATDIM + ncol];
        acc = __builtin_amdgcn_wmma_f32_16x16x4_f32(
            /*neg_a=*/false, a, /*neg_b=*/false, b,
            /*c_mod=*/(short)0, acc, /*reuse_a=*/false, /*reuse_b=*/false);
    }

    // C/D layout: VGPR r -> M=r (lanes 0-15) / M=8+r (lanes 16-31), N=l15
    const int mbase = mtile * 16 + ((lane >> 4) << 3);
#pragma unroll
    for (int r = 0; r < 8; ++r)
        O[(size_t)(mbase + r) * LATDIM + ncol] = acc[r];
}

// ---------------------------------------------------------------------------
// Kernel 2: per-edge scores. One wave32 per edge; lane l handles elements
// 4l..4l+3 of the 128-float Q/K rows (coalesced float4 loads; Q/K are
// L2-resident at 25.6MB each). Quad shuffle-reduce -> one head per quad.
// expAtt written straight into d_out's attn slot; atomicAdd into norm.
// ---------------------------------------------------------------------------
__global__ __launch_bounds__(256) void edge_scores_kernel(
    const float* __restrict__ Q, const float* __restrict__ K,
    const int* __restrict__ rows, const int* __restrict__ cols,
    float* __restrict__ expAtt, float* __restrict__ norm, int E) {

    const int gid  = blockIdx.x * blockDim.x + threadIdx.x;
    const int e    = gid >> 5;
    const int lane = gid & 31;
    if (e >= E) return;

    const int r = rows[e];
    const int c = cols[e];
    const float4 q = *(const float4*)(Q + (size_t)r * LATDIM + lane * 4);
    const float4 k = *(const float4*)(K + (size_t)c * LATDIM + lane * 4);

    float s = q.x * k.x + q.y * k.y + q.z * k.z + q.w * k.w;
    // reduce the 4 lanes of each quad (16 elements per head)
    s += __shfl_xor(s, 1, 32);
    s += __shfl_xor(s, 2, 32);

    s = fminf(10.0f, fmaxf(-10.0f, s));
    const float ex = expf(s);

    if ((lane & 3) == 0) {
        const int h = lane >> 2;
        expAtt[(size_t)e * HEADS + h] = ex;
        atomicAdd(&norm[(size_t)r * HEADS + h], ex);
    }
}

// ---------------------------------------------------------------------------
// Kernel 3: normalize + weighted scatter. One wave per edge. attn buffer is
// read (expAtt) then overwritten (normalized) within the same wave's program
// order -- each element touched only by its own edge's wave.
// ---------------------------------------------------------------------------
__global__ __launch_bounds__(256) void edge_scatter_kernel(
    const float* __restrict__ V,
    const int* __restrict__ rows, const int* __restrict__ cols,
    float* __restrict__ attn, const float* __restrict__ norm,
    float* __restrict__ res, int E) {

    const int gid  = blockIdx.x * blockDim.x + threadIdx.x;
    const int e    = gid >> 5;
    const int lane = gid & 31;
    if (e >= E) return;

    const int r = rows[e];
    const int c = cols[e];
    const int h = lane >> 2;

    const float ex  = attn[(size_t)e * HEADS + h];   // exp values from kernel 2
    const float nrm = norm[(size_t)r * HEADS + h];
    const float a   = ex / (nrm + 1e-8f);

    const float4 v = *(const float4*)(V + (size_t)c * LATDIM + lane * 4);
    float* dst = res + (size_t)r * LATDIM + lane * 4;
    atomicAdd(dst + 0, a * v.x);
    atomicAdd(dst + 1, a * v.y);
    atomicAdd(dst + 2, a * v.z);
    atomicAdd(dst + 3, a * v.w);

    if ((lane & 3) == 0) attn[(size_t)e * HEADS + h] = a;
}

// ---------------------------------------------------------------------------
extern "C" void kernel_launch(void* const* d_in, const int* in_sizes, int n_in,
                              void* d_out, int out_size, void* d_ws, size_t ws_size,
                              hipStream_t stream) {
    const float* embeds = (const float*)d_in[0];
    const float* Wq     = (const float*)d_in[1];
    const float* Wk     = (const float*)d_in[2];
    const float* Wv     = (const float*)d_in[3];
    const int*   rows   = (const int*)d_in[4];
    const int*   cols   = (const int*)d_in[5];

    const int N = in_sizes[0] / LATDIM;   // 50000
    const int E = in_sizes[4];            // 800000

    // Workspace layout (floats): Q | K | V | norm   (~78.4 MB)
    float* ws   = (float*)d_ws;
    float* Q    = ws;
    float* K    = Q + (size_t)N * LATDIM;
    float* Vv   = K + (size_t)N * LATDIM;
    float* norm = Vv + (size_t)N * LATDIM;

    // Output layout: resEmbeds [N,128] then attn [E,8]
    float* res  = (float*)d_out;
    float* attn = res + (size_t)N * LATDIM;

    zero_accum_kernel<<<1024, 256, 0, stream>>>(res, norm, N * LATDIM, N * HEADS);

    dim3 gemm_grid((N + 15) / 16, 3);
    qkv_gemm_kernel<<<gemm_grid, 256, 0, stream>>>(embeds, Wq, Wk, Wv, ws, N);

    const int edge_blocks = (E * 32 + 255) / 256;  // one wave32 per edge
    edge_scores_kernel<<<edge_blocks, 256, 0, stream>>>(Q, K, rows, cols, attn, norm, E);
    edge_scatter_kernel<<<edge_blocks, 256, 0, stream>>>(Vv, rows, cols, attn, norm, res, E);
}